// GeometricReconstructionLoss_67070209294708
// MI455X (gfx1250) — compile-verified
//
#include <hip/hip_runtime.h>

typedef __attribute__((ext_vector_type(2))) float v2f;
typedef __attribute__((ext_vector_type(8))) float v8f;

#define B_      4
#define T_      28
#define N_      1024
#define BT_     (B_ * T_)
#define THREADS 256
#define WAVES   (THREADS / 32)

__device__ __forceinline__ float smooth_l1(float x, float y) {
    float d = fabsf(x - y);
    return d < 1.0f ? 0.5f * d * d : d - 0.5f;
}

// One workgroup per (b,t). Computes:
//   ws[2*bt]   = weights[bt] * sum_{n,d} smoothL1(X[n], Tnearest[n]) / (B*N*3)
//   ws[2*bt+1] = sum_d smoothL1(mean(X)[d], mean(T)[d]) / (B*3)
//
// Nearest-neighbor trick: pad K from 3 to 4 with X[3]=1, T[3]=-0.5*||t||^2 so a
// single V_WMMA_F32_16X16X4_F32 yields d = <p,t> - 0.5*||t||^2, and
// argmin_m dist(n,m) == argmax_m d(n,m)   (dist = ||p||^2 - 2 d).
__global__ __launch_bounds__(THREADS)
void chamfer_bt_kernel(const float* __restrict__ X,
                       const float* __restrict__ Tg,
                       const float* __restrict__ W,
                       float* __restrict__ ws) {
    __shared__ float sX[N_ * 4];     // pred points:   {x,y,z, 1.0}
    __shared__ float sT[N_ * 4];     // target points: {x,y,z, -0.5*||t||^2}
    __shared__ float wpart[WAVES][8];

    const int bt   = blockIdx.x;
    const int tid  = threadIdx.x;
    const int lane = tid & 31;
    const int wave = tid >> 5;
    const int n    = lane & 15;      // row (A) / column (B,D) index within tile
    const int hi   = lane >> 4;      // K-half selector for A/B, M-half for C/D

    const float* Xp = X  + (size_t)bt * (N_ * 3);
    const float* Tp = Tg + (size_t)bt * (N_ * 3);

    // ---- stage into LDS, accumulate centroid partial sums ----
    float cx0 = 0.f, cx1 = 0.f, cx2 = 0.f;
    float ct0 = 0.f, ct1 = 0.f, ct2 = 0.f;
    for (int p = tid; p < N_; p += THREADS) {
        float x0 = Xp[p * 3 + 0], x1 = Xp[p * 3 + 1], x2 = Xp[p * 3 + 2];
        sX[p * 4 + 0] = x0; sX[p * 4 + 1] = x1;
        sX[p * 4 + 2] = x2; sX[p * 4 + 3] = 1.0f;
        cx0 += x0; cx1 += x1; cx2 += x2;
        float t0 = Tp[p * 3 + 0], t1 = Tp[p * 3 + 1], t2 = Tp[p * 3 + 2];
        sT[p * 4 + 0] = t0; sT[p * 4 + 1] = t1;
        sT[p * 4 + 2] = t2;
        sT[p * 4 + 3] = -0.5f * (t0 * t0 + t1 * t1 + t2 * t2);
        ct0 += t0; ct1 += t1; ct2 += t2;
    }
    __syncthreads();

    float accL = 0.f;

    // ---- 8 row-tiles per wave; scan 64 col-tiles with WMMA f32 16x16x4 ----
    #pragma unroll 1
    for (int j = 0; j < 8; ++j) {
        const int r0 = (wave * 8 + j) * 16;
        // A fragment (16x4 f32): lanes 0-15 hold K=0,1; lanes 16-31 hold K=2,3
        v2f a = *(const v2f*)&sX[(r0 + n) * 4 + 2 * hi];

        float best[8];
        int   bidx[8];
        #pragma unroll
        for (int v = 0; v < 8; ++v) { best[v] = -3.0e38f; bidx[v] = 0; }

        // software-pipelined column scan: load B(ct+1) while selecting on d(ct)
        v2f b = *(const v2f*)&sT[n * 4 + 2 * hi];
        #pragma unroll 4
        for (int ct = 0; ct < 64; ++ct) {
            const int c0 = ct * 16;
            v8f acc = {};
            v8f d = __builtin_amdgcn_wmma_f32_16x16x4_f32(
                false, a, false, b, (short)0, acc, false, false);
            const int cn = ((ct + 1) & 63) * 16;
            v2f bn = *(const v2f*)&sT[(cn + n) * 4 + 2 * hi];
            #pragma unroll
            for (int v = 0; v < 8; ++v) {
                bool gt = d[v] > best[v];     // argmax d == argmin dist
                best[v] = gt ? d[v] : best[v];
                bidx[v] = gt ? (c0 + n) : bidx[v];
            }
            b = bn;
        }

        // argmax reduce across the 16-lane half (xor masks stay within halves);
        // tie-break on smaller index to match jnp.argmin first-occurrence.
        #pragma unroll
        for (int mask = 1; mask < 16; mask <<= 1) {
            #pragma unroll
            for (int v = 0; v < 8; ++v) {
                float ob = __shfl_xor(best[v], mask, 32);
                int   oi = __shfl_xor(bidx[v], mask, 32);
                bool take = (ob > best[v]) || (ob == best[v] && oi < bidx[v]);
                best[v] = take ? ob : best[v];
                bidx[v] = take ? oi : bidx[v];
            }
        }

        // lanes with n<8 each handle one row of this tile-half
        if (n < 8) {
            int idx = bidx[0];
            #pragma unroll
            for (int v = 1; v < 8; ++v) idx = (n == v) ? bidx[v] : idx;
            const int row = r0 + 8 * hi + n;
            accL += smooth_l1(sX[row * 4 + 0], sT[idx * 4 + 0])
                  + smooth_l1(sX[row * 4 + 1], sT[idx * 4 + 1])
                  + smooth_l1(sX[row * 4 + 2], sT[idx * 4 + 2]);
        }
    }

    // ---- deterministic wave + block reduction of 7 accumulators ----
    float vals[7] = {accL, cx0, cx1, cx2, ct0, ct1, ct2};
    #pragma unroll
    for (int i = 0; i < 7; ++i) {
        #pragma unroll
        for (int mask = 16; mask >= 1; mask >>= 1)
            vals[i] += __shfl_xor(vals[i], mask, 32);
    }
    if (lane == 0) {
        #pragma unroll
        for (int i = 0; i < 7; ++i) wpart[wave][i] = vals[i];
    }
    __syncthreads();

    if (tid == 0) {
        float tot[7] = {0.f, 0.f, 0.f, 0.f, 0.f, 0.f, 0.f};
        for (int w = 0; w < WAVES; ++w) {
            #pragma unroll
            for (int i = 0; i < 7; ++i) tot[i] += wpart[w][i];
        }
        const float wgt = W[bt];
        ws[2 * bt + 0] = tot[0] * wgt * (1.0f / (float)(B_ * N_ * 3));
        const float invN = 1.0f / (float)N_;
        float lc = smooth_l1(tot[1] * invN, tot[4] * invN)
                 + smooth_l1(tot[2] * invN, tot[5] * invN)
                 + smooth_l1(tot[3] * invN, tot[6] * invN);
        ws[2 * bt + 1] = lc * (1.0f / (float)(B_ * 3));
    }
}

// Fixed-order final sum over 112 per-(b,t) partials -> deterministic scalars.
__global__ void finalize_kernel(const float* __restrict__ ws,
                                float* __restrict__ out) {
    if (threadIdx.x == 0 && blockIdx.x == 0) {
        float l = 0.f, lc = 0.f;
        for (int i = 0; i < BT_; ++i) {
            l  += ws[2 * i + 0];
            lc += ws[2 * i + 1];
        }
        out[0] = l;
        out[1] = lc;
    }
}

extern "C" void kernel_launch(void* const* d_in, const int* in_sizes, int n_in,
                              void* d_out, int out_size, void* d_ws, size_t ws_size,
                              hipStream_t stream) {
    const float* X  = (const float*)d_in[0];   // [B,T,N,3] f32
    const float* Tg = (const float*)d_in[1];   // [B,T,N,3] f32
    const float* W  = (const float*)d_in[2];   // [B,T]     f32
    float* ws  = (float*)d_ws;                 // 2*BT_ floats of scratch
    float* out = (float*)d_out;                // 2 floats: (loss, lossc)

    chamfer_bt_kernel<<<BT_, THREADS, 0, stream>>>(X, Tg, W, ws);
    finalize_kernel<<<1, 32, 0, stream>>>(ws, out);
}